// KalmanFilterSmoother_69776038691039
// MI455X (gfx1250) — compile-verified
//
#include <hip/hip_runtime.h>
#include <hip/hip_bf16.h>

// ---------------------------------------------------------------------------
// Kalman filter + RTS smoother, d=64, sequential scan over N steps.
// Single persistent workgroup (256 threads = 8 wave32) on one WGP.
// All 64x64 matmuls via V_WMMA_F32_16X16X4_F32; state lives in LDS.
// ---------------------------------------------------------------------------

#define ST 65   // LDS row stride (floats): 64+1 -> bank-conflict-free column walks

typedef float v2f __attribute__((ext_vector_type(2)));
typedef float v8f __attribute__((ext_vector_type(8)));

// C(64x64, LDS stride ST) = A(64x64, lda) * op(B)(64x64, ldb) [+ ADD(ldd)]
// op(B) = B if !TRB else B^T.  Executed by all 256 threads; 16 tiles of 16x16,
// 2 tiles per wave, K walked in steps of 4 with v_wmma_f32_16x16x4_f32.
template <bool TRB>
__device__ __forceinline__ void mm64(float* __restrict__ C,
                                     const float* __restrict__ A, int lda,
                                     const float* __restrict__ B, int ldb,
                                     const float* __restrict__ ADD, int ldd,
                                     int tid) {
  const int wave = tid >> 5;
  const int lane = tid & 31;
  const int lh   = lane >> 4;   // lane half: selects K sub-pair / M sub-block
  const int ll   = lane & 15;

#pragma unroll
  for (int t = 0; t < 2; ++t) {
    const int tile = wave + t * 8;      // 0..15
    const int ti = tile >> 2;           // row block
    const int tj = tile & 3;            // col block
    const int m = ti * 16 + ll;         // A row for this lane
    const int n = tj * 16 + ll;         // B col for this lane

    v8f acc;
    if (ADD) {
#pragma unroll
      for (int j = 0; j < 8; ++j)
        acc[j] = ADD[(ti * 16 + (lh << 3) + j) * ldd + tj * 16 + ll];
    } else {
#pragma unroll
      for (int j = 0; j < 8; ++j) acc[j] = 0.0f;
    }

#pragma unroll
    for (int kk = 0; kk < 16; ++kk) {
      const int k0 = kk * 4 + (lh << 1);  // this lane-half holds K = k0, k0+1
      v2f a, b;
      a.x = A[m * lda + k0];
      a.y = A[m * lda + k0 + 1];
      if (TRB) {
        b.x = B[n * ldb + k0];
        b.y = B[n * ldb + k0 + 1];
      } else {
        b.x = B[(k0) * ldb + n];
        b.y = B[(k0 + 1) * ldb + n];
      }
      acc = __builtin_amdgcn_wmma_f32_16x16x4_f32(
          /*neg_a=*/false, a, /*neg_b=*/false, b,
          /*c_mod=*/(short)0, acc, /*reuse_a=*/false, /*reuse_b=*/false);
    }

#pragma unroll
    for (int j = 0; j < 8; ++j)
      C[(ti * 16 + (lh << 3) + j) * ST + tj * 16 + ll] = acc[j];
  }
}

// L = chol(S), lower triangular, cooperative right-looking. Includes barriers.
__device__ __forceinline__ void chol64(float* __restrict__ L,
                                       const float* __restrict__ S, int tid) {
  for (int i = tid; i < 4096; i += 256) {
    const int r = i >> 6, c = i & 63;
    L[r * ST + c] = S[r * ST + c];
  }
  __syncthreads();
  for (int j = 0; j < 64; ++j) {
    if (tid == 0) L[j * ST + j] = sqrtf(L[j * ST + j]);
    __syncthreads();
    const float dinv = 1.0f / L[j * ST + j];
    for (int i = j + 1 + tid; i < 64; i += 256) L[i * ST + j] *= dinv;
    __syncthreads();
    for (int i = j + 1 + tid; i < 64; i += 256) {
      const float lij = L[i * ST + j];
      for (int k = j + 1; k <= i; ++k) L[i * ST + k] -= lij * L[k * ST + j];
    }
    __syncthreads();
  }
}

// In-place solve of (L L^T) X = Z ; one RHS column per thread (64 active).
__device__ __forceinline__ void spdsolve64(const float* __restrict__ L,
                                           float* __restrict__ Z, int tid) {
  if (tid < 64) {
    const int c = tid;
    for (int i = 0; i < 64; ++i) {            // L y = b
      float s = Z[i * ST + c];
      for (int j = 0; j < i; ++j) s -= L[i * ST + j] * Z[j * ST + c];
      Z[i * ST + c] = s / L[i * ST + i];
    }
  }
  __syncthreads();
  if (tid < 64) {
    const int c = tid;
    for (int i = 63; i >= 0; --i) {           // L^T x = y
      float s = Z[i * ST + c];
      for (int j = i + 1; j < 64; ++j) s -= L[j * ST + i] * Z[j * ST + c];
      Z[i * ST + c] = s / L[i * ST + i];
    }
  }
  __syncthreads();
}

__device__ __forceinline__ float dotrow(const float* __restrict__ M, int ld,
                                        int r, const float* __restrict__ v) {
  float s = 0.0f;
#pragma unroll 8
  for (int j = 0; j < 64; ++j) s = __builtin_fmaf(M[r * ld + j], v[j], s);
  return s;
}

__global__ void __launch_bounds__(256, 1)
KalmanFilterSmoother_69776038691039_kernel(
    const float* __restrict__ x0, const float* __restrict__ P0,
    const float* __restrict__ Y,  const float* __restrict__ F,
    const float* __restrict__ Q,  const float* __restrict__ H,
    const float* __restrict__ R,
    float* __restrict__ out_fx, float* __restrict__ out_fP,
    float* __restrict__ out_sx, float* __restrict__ out_sP, int N) {
  // 7 x 64x65 f32 matrices = 116,480 B of LDS (<< 320 KB/WGP)
  __shared__ float sP [64 * ST];   // state covariance (predicted) / P_next
  __shared__ float sA [64 * ST];   // HP, then Pf (fwd) ; P_n (bwd)
  __shared__ float sS [64 * ST];   // S (fwd) ; P_fc (bwd)
  __shared__ float sL [64 * ST];   // Cholesky factor
  __shared__ float sK [64 * ST];   // Kt (fwd) ; Jt (bwd)
  __shared__ float sT1[64 * ST];
  __shared__ float sT2[64 * ST];
  __shared__ float svx[64], sv1[64], svf[64], svn[64];

  const int tid = threadIdx.x;

  for (int i = tid; i < 4096; i += 256)
    sP[(i >> 6) * ST + (i & 63)] = P0[i];
  if (tid < 64) svx[tid] = x0[tid];
  __syncthreads();

  // ------------------------------ forward ---------------------------------
  for (int n = 0; n < N; ++n) {
    mm64<false>(sA, H, 64, sP, ST, nullptr, 0, tid);        // HP = H @ P
    __syncthreads();
    mm64<true >(sS, sA, ST, H, 64, R, 64, tid);             // S = HP @ H^T + R
    __syncthreads();
    for (int i = tid; i < 4096; i += 256) {                 // Kt <- HP
      const int r = i >> 6, c = i & 63;
      sK[r * ST + c] = sA[r * ST + c];
    }
    __syncthreads();
    chol64(sL, sS, tid);                                    // L = chol(S)
    spdsolve64(sL, sK, tid);                                // Kt = S^-1 H P
    if (tid < 64)                                           // innov = y - H x
      sv1[tid] = Y[(size_t)n * 64 + tid] - dotrow(H, 64, tid, svx);
    __syncthreads();
    if (tid < 64) {                                         // xf = x + Kt innov
      const float v = svx[tid] + dotrow(sK, ST, tid, sv1);
      svf[tid] = v;
      out_fx[(size_t)n * 64 + tid] = v;
    }
    __syncthreads();
    mm64<true >(sT1, sK, ST, sS, ST, nullptr, 0, tid);      // T1 = Kt @ S^T
    __syncthreads();
    mm64<false>(sT2, sT1, ST, sK, ST, nullptr, 0, tid);     // T2 = T1 @ Kt
    __syncthreads();
    for (int i = tid; i < 4096; i += 256) {                 // Pf = P - T2
      const int r = i >> 6, c = i & 63;
      const float v = sP[r * ST + c] - sT2[r * ST + c];
      sA[r * ST + c] = v;
      out_fP[(size_t)n * 4096 + i] = v;
    }
    __syncthreads();
    mm64<false>(sT1, F, 64, sA, ST, nullptr, 0, tid);       // T1 = F @ Pf
    __syncthreads();
    mm64<true >(sP, sT1, ST, F, 64, Q, 64, tid);            // P = T1 @ F^T + Q
    if (tid < 64) svx[tid] = dotrow(F, 64, tid, svf);       // x = F @ xf
    __syncthreads();
  }

  // --------------------------- backward (RTS) -----------------------------
  for (int i = tid; i < 4096; i += 256) {
    const float v = out_fP[(size_t)(N - 1) * 4096 + i];
    sP[(i >> 6) * ST + (i & 63)] = v;
    out_sP[(size_t)(N - 1) * 4096 + i] = v;
  }
  if (tid < 64) {
    const float v = out_fx[(size_t)(N - 1) * 64 + tid];
    svn[tid] = v;
    out_sx[(size_t)(N - 1) * 64 + tid] = v;
  }
  __syncthreads();

  for (int n = N - 2; n >= 0; --n) {
    for (int i = tid; i < 4096; i += 256)                   // load P_n
      sA[(i >> 6) * ST + (i & 63)] = out_fP[(size_t)n * 4096 + i];
    if (tid < 64) svx[tid] = out_fx[(size_t)n * 64 + tid];
    if (n > 0) {                                            // hide next reload
      __builtin_prefetch(&out_fP[(size_t)(n - 1) * 4096 + tid * 16], 0, 0);
      if (tid == 0) __builtin_prefetch(&out_fx[(size_t)(n - 1) * 64], 0, 0);
    }
    __syncthreads();
    mm64<false>(sT1, F, 64, sA, ST, nullptr, 0, tid);       // T1 = F @ P_n
    __syncthreads();
    mm64<true >(sS, sT1, ST, F, 64, Q, 64, tid);            // P_fc = T1 F^T + Q
    __syncthreads();
    for (int i = tid; i < 4096; i += 256) {                 // Jt <- F @ P_n
      const int r = i >> 6, c = i & 63;
      sK[r * ST + c] = sT1[r * ST + c];
    }
    __syncthreads();
    chol64(sL, sS, tid);                                    // L = chol(P_fc)
    spdsolve64(sL, sK, tid);                                // Jt = P_fc^-1 F P_n
    if (tid < 64)                                           // v1 = x_next - F x_n
      sv1[tid] = svn[tid] - dotrow(F, 64, tid, svx);
    __syncthreads();
    if (tid < 64) {                                         // xs = x_n + Jt v1
      const float v = svx[tid] + dotrow(sK, ST, tid, sv1);
      out_sx[(size_t)n * 64 + tid] = v;
      svn[tid] = v;
    }
    for (int i = tid; i < 4096; i += 256) {                 // D = P_next - P_fc
      const int r = i >> 6, c = i & 63;
      sT2[r * ST + c] = sP[r * ST + c] - sS[r * ST + c];
    }
    __syncthreads();
    mm64<false>(sT1, sK, ST, sT2, ST, nullptr, 0, tid);     // T1 = Jt @ D
    __syncthreads();
    mm64<true >(sT2, sT1, ST, sK, ST, sA, ST, tid);         // Ps = P_n + T1 Jt^T
    __syncthreads();
    for (int i = tid; i < 4096; i += 256) {                 // commit P_next = Ps
      const int r = i >> 6, c = i & 63;
      const float v = sT2[r * ST + c];
      sP[r * ST + c] = v;
      out_sP[(size_t)n * 4096 + i] = v;
    }
    __syncthreads();
  }
}

extern "C" void kernel_launch(void* const* d_in, const int* in_sizes, int n_in,
                              void* d_out, int out_size, void* d_ws, size_t ws_size,
                              hipStream_t stream) {
  const float* x0 = (const float*)d_in[0];
  const float* P0 = (const float*)d_in[1];
  const float* Y  = (const float*)d_in[2];
  const float* F  = (const float*)d_in[3];
  const float* Q  = (const float*)d_in[4];
  const float* H  = (const float*)d_in[5];
  const float* R  = (const float*)d_in[6];

  const int N = in_sizes[2] / 64;

  float* out = (float*)d_out;   // (fx, fP, sm_means, sm_covs) concatenated
  float* fx = out;
  float* fP = fx + (size_t)N * 64;
  float* sx = fP + (size_t)N * 4096;
  float* sP = sx + (size_t)N * 64;

  KalmanFilterSmoother_69776038691039_kernel<<<dim3(1), dim3(256), 0, stream>>>(
      x0, P0, Y, F, Q, H, R, fx, fP, sx, sP, N);
}